// GAT_66838281060989
// MI455X (gfx1250) — compile-verified
//
#include <hip/hip_runtime.h>
#include <hip/hip_bf16.h>
#include <math.h>

typedef __attribute__((ext_vector_type(2))) float v2f;
typedef __attribute__((ext_vector_type(8))) float v8f;

#define EPSLN 1e-5f

// ---------------------------------------------------------------------------
// fp32 WMMA: D(16x16) = A(16x4) * B(4x16) + C   (V_WMMA_F32_16X16X4_F32)
// ---------------------------------------------------------------------------
__device__ __forceinline__ v8f wmma_f32_4(v2f a, v2f b, v8f c) {
    return __builtin_amdgcn_wmma_f32_16x16x4_f32(
        /*neg_a=*/false, a, /*neg_b=*/false, b,
        /*c_mod=*/(short)0, c, /*reuse_a=*/false, /*reuse_b=*/false);
}

__device__ __forceinline__ void atomicMaxF(float* addr, float val) {
    if (val >= 0.0f) atomicMax((int*)addr, __float_as_int(val));
    else             atomicMin((unsigned int*)addr, __float_as_uint(val));
}

// ---------------------------------------------------------------------------
// Kernel 1: fused Q/K/V GEMM.  feat[N,128] x W[128,128] -> out[N,128] (x3)
// One block = 64 rows (4 M-tiles).  8 waves; wave w owns column tile w.
// B fragments are register-cached across the 4 M-tiles (4x fewer vmem loads).
// ---------------------------------------------------------------------------
__global__ __launch_bounds__(256) void qkv_kernel(
    const float* __restrict__ feat,
    const float* __restrict__ Wq, const float* __restrict__ Wk,
    const float* __restrict__ Wv,
    float* __restrict__ q, float* __restrict__ k, float* __restrict__ v, int n)
{
    __shared__ float sA[64][132];
    const int row0 = blockIdx.x * 64;
    const int tid  = threadIdx.x;

    for (int i = tid; i < 64 * 32; i += 256) {          // float4 granularity
        int r = i >> 5, c4 = i & 31;
        int gr = row0 + r; if (gr >= n) gr = n - 1;
        ((float4*)&sA[r][0])[c4] = ((const float4*)(feat + (size_t)gr * 128))[c4];
    }
    __syncthreads();

    const int wave = tid >> 5;
    const int lane = tid & 31;
    const int half = lane >> 4;
    const int lr   = lane & 15;
    const int col  = wave * 16 + lr;

    const float* Ws[3] = { Wq, Wk, Wv };
    float*       Os[3] = { q,  k,  v  };

#pragma unroll
    for (int m = 0; m < 3; ++m) {
        const float* __restrict__ W = Ws[m];
        v2f bfr[32];
#pragma unroll
        for (int kq = 0; kq < 32; ++kq) {
            const int kk = 4 * kq + 2 * half;
            bfr[kq] = (v2f){ W[(size_t)kk * 128 + col],
                             W[(size_t)(kk + 1) * 128 + col] };
        }
        float* __restrict__ O = Os[m];
#pragma unroll
        for (int mt = 0; mt < 4; ++mt) {
            v8f acc = {};
#pragma unroll
            for (int kq = 0; kq < 32; ++kq) {
                const int kk = 4 * kq + 2 * half;
                v2f a = { sA[mt * 16 + lr][kk], sA[mt * 16 + lr][kk + 1] };
                acc = wmma_f32_4(a, bfr[kq], acc);
            }
#pragma unroll
            for (int r = 0; r < 8; ++r) {
                int gr = row0 + mt * 16 + r + 8 * half;
                if (gr < n) O[(size_t)gr * 128 + col] = acc[r];
            }
        }
    }
}

// ---------------------------------------------------------------------------
// Kernel 2: init smax=-inf, ssum=0, ft2=0
// ---------------------------------------------------------------------------
__global__ __launch_bounds__(256) void init_kernel(
    float* __restrict__ smax, float* __restrict__ ssum,
    float* __restrict__ ft2, int n)
{
    int i = blockIdx.x * 256 + threadIdx.x;
    if (i < n * 8) { smax[i] = -INFINITY; ssum[i] = 0.0f; }
    if (i < n * 128) ft2[i] = 0.0f;
}

// ---------------------------------------------------------------------------
// Kernel 3: per-(edge,head) score + atomic max over dst
// ---------------------------------------------------------------------------
__global__ __launch_bounds__(256) void scores_kernel(
    const float* __restrict__ kbuf, const float* __restrict__ qbuf,
    const int* __restrict__ src, const int* __restrict__ dst,
    float* __restrict__ scores, float* __restrict__ smax, int eh)
{
    int t = blockIdx.x * 256 + threadIdx.x;
    if (t >= eh) return;
    const int e = t >> 3, h = t & 7;
    const int s = src[e], d = dst[e];
    const float4* kp = (const float4*)(kbuf + (size_t)s * 128 + h * 16);
    const float4* qp = (const float4*)(qbuf + (size_t)d * 128 + h * 16);
    float acc = 0.0f;
#pragma unroll
    for (int i = 0; i < 4; ++i) {
        float4 a = kp[i], b = qp[i];
        acc += a.x * b.x + a.y * b.y + a.z * b.z + a.w * b.w;
    }
    acc *= 0.08838834764831845f;   // 1/sqrt(DH*H) = 1/sqrt(128)
    scores[t] = acc;
    atomicMaxF(&smax[d * 8 + h], acc);
}

// ---------------------------------------------------------------------------
// Kernel 4: exp(s - max) and atomic segment sum
// ---------------------------------------------------------------------------
__global__ __launch_bounds__(256) void expsum_kernel(
    const int* __restrict__ dst, float* __restrict__ scores,
    const float* __restrict__ smax, float* __restrict__ ssum, int eh)
{
    int t = blockIdx.x * 256 + threadIdx.x;
    if (t >= eh) return;
    const int e = t >> 3, h = t & 7;
    const int d = dst[e];
    float ex = expf(scores[t] - smax[d * 8 + h]);
    scores[t] = ex;
    atomicAdd(&ssum[d * 8 + h], ex);
}

// ---------------------------------------------------------------------------
// Kernel 5: normalize + weighted scatter of v[src] into ft2[dst]
// ---------------------------------------------------------------------------
__global__ __launch_bounds__(256) void scatter_kernel(
    const int* __restrict__ src, const int* __restrict__ dst,
    const float* __restrict__ scores, const float* __restrict__ ssum,
    const float* __restrict__ vbuf, float* __restrict__ ft2, int eh)
{
    int t = blockIdx.x * 256 + threadIdx.x;
    if (t >= eh) return;
    const int e = t >> 3, h = t & 7;
    const int s = src[e], d = dst[e];
    const float a = scores[t] / ssum[d * 8 + h];
    const float* vp = vbuf + (size_t)s * 128 + h * 16;
    float*       fp = ft2  + (size_t)d * 128 + h * 16;
#pragma unroll
    for (int i = 0; i < 16; ++i)
        atomicAdd(&fp[i], vp[i] * a);
}

// ---------------------------------------------------------------------------
// Kernel 6: rst = LayerNorm(ft2 + feat).  16 rows / block, 16 lanes / row.
// ---------------------------------------------------------------------------
__global__ __launch_bounds__(256) void ln1_kernel(
    const float* __restrict__ ft2, const float* __restrict__ feat,
    const float* __restrict__ g, const float* __restrict__ b,
    float* __restrict__ rst, int n)
{
    const int row = blockIdx.x * 16 + (threadIdx.x >> 4);
    const int c0  = threadIdx.x & 15;
    if (row >= n) return;
    const float* p1 = ft2  + (size_t)row * 128;
    const float* p2 = feat + (size_t)row * 128;
    float xv[8], s = 0.0f, ss = 0.0f;
#pragma unroll
    for (int j = 0; j < 8; ++j) {
        int c = c0 + 16 * j;
        float x = p1[c] + p2[c];
        xv[j] = x; s += x; ss += x * x;
    }
#pragma unroll
    for (int m = 1; m < 16; m <<= 1) {
        s  += __shfl_xor(s,  m, 32);
        ss += __shfl_xor(ss, m, 32);
    }
    const float mean = s * (1.0f / 128.0f);
    const float var  = ss * (1.0f / 128.0f) - mean * mean;
    const float inv  = rsqrtf(var + EPSLN);
    float* o = rst + (size_t)row * 128;
#pragma unroll
    for (int j = 0; j < 8; ++j) {
        int c = c0 + 16 * j;
        o[c] = (xv[j] - mean) * inv * g[c] + b[c];
    }
}

// ---------------------------------------------------------------------------
// Kernel 7: h = PReLU(rst @ W1 + b1).  [N,128]x[128,512].
// 64 rows / block; 8 waves x 4 col tiles; B register-cached across M-tiles.
// ---------------------------------------------------------------------------
__global__ __launch_bounds__(256) void ffn1_kernel(
    const float* __restrict__ rst, const float* __restrict__ W1,
    const float* __restrict__ b1, const float* __restrict__ pw,
    float* __restrict__ hbuf, int n)
{
    __shared__ float sA[64][132];
    const int row0 = blockIdx.x * 64;
    const int tid  = threadIdx.x;
    for (int i = tid; i < 64 * 32; i += 256) {
        int r = i >> 5, c4 = i & 31;
        int gr = row0 + r; if (gr >= n) gr = n - 1;
        ((float4*)&sA[r][0])[c4] = ((const float4*)(rst + (size_t)gr * 128))[c4];
    }
    __syncthreads();

    const int wave = tid >> 5;
    const int lane = tid & 31;
    const int half = lane >> 4;
    const int lr   = lane & 15;

#pragma unroll
    for (int j = 0; j < 4; ++j) {
        const int col = (wave + 8 * j) * 16 + lr;       // 0..511
        v2f bfr[32];
#pragma unroll
        for (int kq = 0; kq < 32; ++kq) {
            const int kk = 4 * kq + 2 * half;
            bfr[kq] = (v2f){ W1[(size_t)kk * 512 + col],
                             W1[(size_t)(kk + 1) * 512 + col] };
        }
        const float bias = b1[col], slope = pw[col];
#pragma unroll
        for (int mt = 0; mt < 4; ++mt) {
            v8f acc = {};
#pragma unroll
            for (int kq = 0; kq < 32; ++kq) {
                const int kk = 4 * kq + 2 * half;
                v2f a = { sA[mt * 16 + lr][kk], sA[mt * 16 + lr][kk + 1] };
                acc = wmma_f32_4(a, bfr[kq], acc);
            }
#pragma unroll
            for (int r = 0; r < 8; ++r) {
                int gr = row0 + mt * 16 + r + 8 * half;
                if (gr < n) {
                    float h = acc[r] + bias;
                    h = (h >= 0.0f) ? h : slope * h;
                    hbuf[(size_t)gr * 512 + col] = h;
                }
            }
        }
    }
}

// ---------------------------------------------------------------------------
// Kernel 8: out = LayerNorm(rst + h @ W2 + b2).  [N,512]x[512,128].
// 64 rows / block; K chunked 4x128 through a 64x128 LDS A-tile; the same LDS
// tile is reused to stage the residual sum for the fused LayerNorm.
// ---------------------------------------------------------------------------
__global__ __launch_bounds__(256) void ffn2_ln_kernel(
    const float* __restrict__ hbuf, const float* __restrict__ W2,
    const float* __restrict__ b2, const float* __restrict__ rst,
    const float* __restrict__ g, const float* __restrict__ be,
    float* __restrict__ out, int n)
{
    __shared__ float sA[64][132];
    const int row0 = blockIdx.x * 64;
    const int tid  = threadIdx.x;
    const int wave = tid >> 5;
    const int lane = tid & 31;
    const int half = lane >> 4;
    const int lr   = lane & 15;
    const int col  = wave * 16 + lr;     // 0..127

    v8f acc[4] = {};
#pragma unroll
    for (int kc = 0; kc < 4; ++kc) {
        __syncthreads();                 // protect previous chunk / LN reuse
        for (int i = tid; i < 64 * 32; i += 256) {
            int r = i >> 5, c4 = i & 31;
            int gr = row0 + r; if (gr >= n) gr = n - 1;
            ((float4*)&sA[r][0])[c4] =
                ((const float4*)(hbuf + (size_t)gr * 512 + kc * 128))[c4];
        }
        __syncthreads();

        v2f bfr[32];
#pragma unroll
        for (int kq = 0; kq < 32; ++kq) {
            const int kk = kc * 128 + 4 * kq + 2 * half;
            bfr[kq] = (v2f){ W2[(size_t)kk * 128 + col],
                             W2[(size_t)(kk + 1) * 128 + col] };
        }
#pragma unroll
        for (int mt = 0; mt < 4; ++mt) {
#pragma unroll
            for (int kq = 0; kq < 32; ++kq) {
                const int kk = 4 * kq + 2 * half;
                v2f a = { sA[mt * 16 + lr][kk], sA[mt * 16 + lr][kk + 1] };
                acc[mt] = wmma_f32_4(a, bfr[kq], acc[mt]);
            }
        }
    }

    __syncthreads();
    // stage bias + residual into the (now free) LDS tile
    const float bias = b2[col];
#pragma unroll
    for (int mt = 0; mt < 4; ++mt) {
#pragma unroll
        for (int r = 0; r < 8; ++r) {
            const int rl = mt * 16 + r + 8 * half;
            int gr = row0 + rl; if (gr >= n) gr = n - 1;
            sA[rl][col] = acc[mt][r] + bias + rst[(size_t)gr * 128 + col];
        }
    }
    __syncthreads();

    // fused LayerNorm: 4 lanes per row, 64 rows per block
    const int rl2 = tid >> 2;            // 0..63
    const int c0  = tid & 3;
    const int row = row0 + rl2;
    float xv[32], s = 0.0f, ss = 0.0f;
#pragma unroll
    for (int j = 0; j < 32; ++j) {
        int c = c0 + 4 * j;
        float x = sA[rl2][c];
        xv[j] = x; s += x; ss += x * x;
    }
#pragma unroll
    for (int m = 1; m < 4; m <<= 1) {
        s  += __shfl_xor(s,  m, 32);
        ss += __shfl_xor(ss, m, 32);
    }
    const float mean = s * (1.0f / 128.0f);
    const float var  = ss * (1.0f / 128.0f) - mean * mean;
    const float inv  = rsqrtf(var + EPSLN);
    if (row < n) {
        float* o = out + (size_t)row * 128;
#pragma unroll
        for (int j = 0; j < 32; ++j) {
            int c = c0 + 4 * j;
            o[c] = (xv[j] - mean) * inv * g[c] + be[c];
        }
    }
}

// ---------------------------------------------------------------------------
extern "C" void kernel_launch(void* const* d_in, const int* in_sizes, int n_in,
                              void* d_out, int out_size, void* d_ws, size_t ws_size,
                              hipStream_t stream)
{
    const float* feat  = (const float*)d_in[0];
    const int*   src   = (const int*)  d_in[1];
    const int*   dst   = (const int*)  d_in[2];
    const float* Wq    = (const float*)d_in[3];
    const float* Wk    = (const float*)d_in[4];
    const float* Wv    = (const float*)d_in[5];
    const float* ln1_g = (const float*)d_in[6];
    const float* ln1_b = (const float*)d_in[7];
    const float* ln2_g = (const float*)d_in[8];
    const float* ln2_b = (const float*)d_in[9];
    const float* W1    = (const float*)d_in[10];
    const float* b1    = (const float*)d_in[11];
    const float* pw    = (const float*)d_in[12];
    const float* W2    = (const float*)d_in[13];
    const float* b2    = (const float*)d_in[14];

    const int n = in_sizes[0] / 128;     // 40000
    const int e = in_sizes[1];           // 640000
    const int eh = e * 8;

    float* ws = (float*)d_ws;
    size_t off = 0;
    float* q      = ws + off; off += (size_t)n * 128;
    float* kbuf   = ws + off; off += (size_t)n * 128;
    float* vbuf   = ws + off; off += (size_t)n * 128;
    float* ft2    = ws + off; off += (size_t)n * 128;
    float* rst    = ws + off; off += (size_t)n * 128;
    float* hbuf   = ws + off; off += (size_t)n * 512;
    float* scores = ws + off; off += (size_t)eh;
    float* smax   = ws + off; off += (size_t)n * 8;
    float* ssum   = ws + off; off += (size_t)n * 8;
    (void)off; (void)ws_size; (void)n_in; (void)out_size;

    const int NT64 = (n + 63) / 64;             // 64-row GEMM tiles
    const int NT16 = (n + 15) / 16;             // 16-row LN tiles
    const int NB   = (n * 128 + 255) / 256;     // init blocks
    const int EB   = (eh + 255) / 256;          // edge blocks

    qkv_kernel<<<NT64, 256, 0, stream>>>(feat, Wq, Wk, Wv, q, kbuf, vbuf, n);
    init_kernel<<<NB, 256, 0, stream>>>(smax, ssum, ft2, n);
    scores_kernel<<<EB, 256, 0, stream>>>(kbuf, q, src, dst, scores, smax, eh);
    expsum_kernel<<<EB, 256, 0, stream>>>(dst, scores, smax, ssum, eh);
    scatter_kernel<<<EB, 256, 0, stream>>>(src, dst, scores, ssum, vbuf, ft2, eh);
    ln1_kernel<<<NT16, 256, 0, stream>>>(ft2, feat, ln1_g, ln1_b, rst, n);
    ffn1_kernel<<<NT64, 256, 0, stream>>>(rst, W1, b1, pw, hbuf, n);
    ffn2_ln_kernel<<<NT64, 256, 0, stream>>>(hbuf, W2, b2, rst, ln2_g, ln2_b,
                                             (float*)d_out, n);
}